// DeformableCrossAttention_37374805410513
// MI455X (gfx1250) — compile-verified
//
#include <hip/hip_runtime.h>
#include <hip/hip_bf16.h>

#define BS 4
#define NQ 22500
#define DIMS 256
#define H_HEADS 8
#define HEAD_DIM 32
#define NUM_VALUE 29930   // 150*150 + 75*75 + 38*38 + 19*19
#define BLK_M 256         // 8 waves x 32 rows
#define BLK_N 64

typedef __attribute__((ext_vector_type(2))) float v2f;
typedef __attribute__((ext_vector_type(4))) float v4f;
typedef __attribute__((ext_vector_type(8))) float v8f;
typedef int v4i_raw __attribute__((vector_size(4 * sizeof(int))));

#if __has_builtin(__builtin_amdgcn_global_load_async_to_lds_b128)
#define HAVE_ASYNC_LDS 1
#else
#define HAVE_ASYNC_LDS 0
#endif

__device__ __forceinline__ void wait_asynccnt0() {
#if __has_builtin(__builtin_amdgcn_s_wait_asynccnt)
  __builtin_amdgcn_s_wait_asynccnt(0);
#else
  asm volatile("s_wait_asynccnt 0x0" ::: "memory");
#endif
}

// ---------------------------------------------------------------------------
// C[M x N] = A[M x 256] * W[256 x N] + bias[N]   (fp32, V_WMMA_F32_16X16X4_F32)
//
// Block = 256 threads = 8 waves. Block tile 256(M) x 64(N): wave w owns rows
// [blk + 32w, +32) x all 64 cols = 8 accumulator tiles (2 M-subtiles x 4
// N-subtiles). The full 256x64 W slab (64 KB) is staged into LDS once per
// block via global_load_async_to_lds_b128 (ASYNCcnt path), then every wave
// broadcast-reads the same B fragments from LDS each K-step:
//   2 contiguous float2 A loads + 4 LDS B-frags + 8 WMMAs
// (each B fragment feeds both M-subtiles). A-frag per ISA 16x4 f32 layout:
// lanes 0-15 hold K=k0,k0+1; lanes 16-31 hold k0+2,k0+3. B vgpr j holds rows
// k0+2*half+j. EXEC stays all-1s through the WMMA loop (rows clamped,
// stores predicated only in the tail branch).
// ---------------------------------------------------------------------------
__global__ __launch_bounds__(256) void gemm_wmma_f32(
    const float* __restrict__ A, const float* __restrict__ W,
    const float* __restrict__ bias, float* __restrict__ C,
    int M, int N) {
  __shared__ float Bs[256 * BLK_N];   // 64 KB: Bs[k*64 + c] = W[k][tileN0 + c]

  const int tid = threadIdx.x;
  const int lane = tid & 31;
  const int wave = tid >> 5;                      // 0..7 -> 32-row M sub-tile
  const int tileN0 = blockIdx.y * BLK_N;
  const int tileM = blockIdx.x * BLK_M + wave * 32;

  // ---- stage the 256 x 64 slab of W: 4096 16B chunks, 16 per thread -------
#pragma unroll
  for (int i = 0; i < 16; ++i) {
    const int chunk = tid + i * 256;
    const int kr = chunk >> 4;                    // W row (K index)
    const int c4 = (chunk & 15) * 4;              // starting col within slab
    const float* gp = W + (size_t)kr * N + tileN0 + c4;
    float* lp = &Bs[kr * BLK_N + c4];
#if HAVE_ASYNC_LDS
    __builtin_amdgcn_global_load_async_to_lds_b128(
        (__attribute__((address_space(1))) v4i_raw*)(unsigned long long)gp,
        (__attribute__((address_space(3))) v4i_raw*)(unsigned long long)
            (unsigned)(unsigned long long)lp,
        0, 0);
#else
    *(v4f*)lp = *(const v4f*)gp;
#endif
  }
#if HAVE_ASYNC_LDS
  wait_asynccnt0();
#endif
  __syncthreads();

  const int half = lane >> 4;                     // 0: K lo pair, 1: K hi pair
  const int lrow = lane & 15;

  int arow0 = tileM + lrow;
  int arow1 = tileM + 16 + lrow;
  if (arow0 >= M) arow0 = M - 1;                  // clamp; stores are guarded
  if (arow1 >= M) arow1 = M - 1;
  const float* __restrict__ Ap0 = A + (size_t)arow0 * DIMS;
  const float* __restrict__ Ap1 = A + (size_t)arow1 * DIMS;

  v8f acc0[4] = {};
  v8f acc1[4] = {};
#pragma unroll 2
  for (int k0 = 0; k0 < DIMS; k0 += 4) {
    const int ka = k0 + 2 * half;
    v2f a0, a1;
    a0.x = Ap0[ka];
    a0.y = Ap0[ka + 1];
    a1.x = Ap1[ka];
    a1.y = Ap1[ka + 1];
    const float* __restrict__ brow0 = &Bs[ka * BLK_N + lrow];
#pragma unroll
    for (int nt = 0; nt < 4; ++nt) {
      v2f b;
      b.x = brow0[nt * 16];
      b.y = brow0[nt * 16 + BLK_N];
      // (neg_a, A, neg_b, B, c_mod, C, reuse_a, reuse_b)
      acc0[nt] = __builtin_amdgcn_wmma_f32_16x16x4_f32(false, a0, false, b,
                                                       (short)0, acc0[nt],
                                                       false, false);
      acc1[nt] = __builtin_amdgcn_wmma_f32_16x16x4_f32(false, a1, false, b,
                                                       (short)0, acc1[nt],
                                                       false, false);
    }
  }

  // ---- epilogue: bias + store (guard hoisted to a wave-uniform branch) ----
  if (tileM + 32 <= M) {
#pragma unroll
    for (int nt = 0; nt < 4; ++nt) {
      const int col = tileN0 + nt * 16 + lrow;
      const float bv = bias[col];
#pragma unroll
      for (int i = 0; i < 8; ++i) {
        const int r = half * 8 + i;               // C/D vgpr i -> rows i, i+8
        C[(size_t)(tileM + r) * N + col] = acc0[nt][i] + bv;
        C[(size_t)(tileM + 16 + r) * N + col] = acc1[nt][i] + bv;
      }
    }
  } else {
#pragma unroll
    for (int nt = 0; nt < 4; ++nt) {
      const int col = tileN0 + nt * 16 + lrow;
      const float bv = bias[col];
#pragma unroll
      for (int i = 0; i < 8; ++i) {
        const int r0 = tileM + half * 8 + i;
        const int r1 = r0 + 16;
        if (r0 < M) C[(size_t)r0 * N + col] = acc0[nt][i] + bv;
        if (r1 < M) C[(size_t)r1 * N + col] = acc1[nt][i] + bv;
      }
    }
  }
}

// ---------------------------------------------------------------------------
// In-place softmax over contiguous groups of 16 (the L*P axis per (b,q,head)).
// ---------------------------------------------------------------------------
__global__ __launch_bounds__(256) void softmax16_kernel(float* __restrict__ aw,
                                                        int total) {
  const int t = blockIdx.x * blockDim.x + threadIdx.x;
  if (t >= total) return;
  float* __restrict__ p = aw + (size_t)t * 16;
  float v[16];
  float m = -3.4e38f;
#pragma unroll
  for (int i = 0; i < 16; ++i) { v[i] = p[i]; m = fmaxf(m, v[i]); }
  float s = 0.f;
#pragma unroll
  for (int i = 0; i < 16; ++i) { v[i] = __expf(v[i] - m); s += v[i]; }
  const float inv = 1.f / s;
#pragma unroll
  for (int i = 0; i < 16; ++i) p[i] = v[i] * inv;
}

// ---------------------------------------------------------------------------
// Bilinear tap: clamped load always executes (no divergence), validity folded
// into the weight. vl points at v[b, level_start, h*32 + d].
// ---------------------------------------------------------------------------
__device__ __forceinline__ float bilin_tap(const float* __restrict__ vl,
                                           int xi, int yi, int Wd, int Hd,
                                           float wgt) {
  const int xc = min(max(xi, 0), Wd - 1);
  const int yc = min(max(yi, 0), Hd - 1);
  const float g = vl[(size_t)(yc * Wd + xc) * DIMS];
  const bool ok = (xi >= 0) && (xi < Wd) && (yi >= 0) && (yi < Hd);
  return ok ? g * wgt : 0.f;
}

// ---------------------------------------------------------------------------
// One wave32 per (b, q, head): lane d owns channel d (HEAD_DIM == 32 == wave
// size). Each tap is a fully-coalesced 128B row load from the projected value.
// ---------------------------------------------------------------------------
__global__ __launch_bounds__(256) void ms_sample_kernel(
    const float* __restrict__ vproj,  // (BS, NUM_VALUE, 256)
    const float* __restrict__ off,    // (BS, NQ, 256) = (H,L,P,2)
    const float* __restrict__ aw,     // (BS, NQ, 128) = (H,L*P), softmaxed
    const float* __restrict__ ref,    // (BS, NQ, 4, 2)
    float* __restrict__ res) {        // (BS, NQ, 256)
  const int d = threadIdx.x & 31;
  const int gw = blockIdx.x * 8 + (threadIdx.x >> 5);
  if (gw >= BS * NQ * H_HEADS) return;
  const int h = gw & 7;
  const int bq = gw >> 3;           // b*NQ + q
  const int b = bq / NQ;

  const int lw[4] = {150, 75, 38, 19};
  const int lh[4] = {150, 75, 38, 19};
  const int ls[4] = {0, 22500, 28125, 29569};

  const float* __restrict__ op = off + (size_t)bq * 256 + h * 32;
  const float* __restrict__ ap = aw + (size_t)bq * 128 + h * 16;
  const float* __restrict__ rp = ref + (size_t)bq * 8;
  const float* __restrict__ vb =
      vproj + (size_t)b * NUM_VALUE * DIMS + h * HEAD_DIM + d;

  float acc = 0.f;
#pragma unroll
  for (int l = 0; l < 4; ++l) {
    const int Wd = lw[l], Hd = lh[l];
    const float fw = (float)Wd, fh = (float)Hd;
    // x = (ref_x + off_x / W) * W - 0.5 = ref_x*W + off_x - 0.5
    const float rx = rp[l * 2 + 0] * fw - 0.5f;
    const float ry = rp[l * 2 + 1] * fh - 0.5f;
    const float* __restrict__ vl = vb + (size_t)ls[l] * DIMS;
#pragma unroll
    for (int p = 0; p < 4; ++p) {
      const float x = rx + op[(l * 4 + p) * 2 + 0];
      const float y = ry + op[(l * 4 + p) * 2 + 1];
      const float a = ap[l * 4 + p];
      const float x0f = floorf(x), y0f = floorf(y);
      const float lx = x - x0f, ly = y - y0f;
      const int x0 = (int)x0f, y0 = (int)y0f;
      acc += bilin_tap(vl, x0,     y0,     Wd, Hd, (1.f - lx) * (1.f - ly) * a);
      acc += bilin_tap(vl, x0 + 1, y0,     Wd, Hd, lx * (1.f - ly) * a);
      acc += bilin_tap(vl, x0,     y0 + 1, Wd, Hd, (1.f - lx) * ly * a);
      acc += bilin_tap(vl, x0 + 1, y0 + 1, Wd, Hd, lx * ly * a);
    }
  }
  res[(size_t)bq * DIMS + h * HEAD_DIM + d] = acc;
}

// ---------------------------------------------------------------------------
extern "C" void kernel_launch(void* const* d_in, const int* in_sizes, int n_in,
                              void* d_out, int out_size, void* d_ws,
                              size_t ws_size, hipStream_t stream) {
  const float* query = (const float*)d_in[0];
  const float* value = (const float*)d_in[1];
  const float* refpt = (const float*)d_in[2];
  // d_in[3] spatial_shapes, d_in[4] level_start_index: compile-time constants
  const float* W_so = (const float*)d_in[5];
  const float* b_so = (const float*)d_in[6];
  const float* W_aw = (const float*)d_in[7];
  const float* b_aw = (const float*)d_in[8];
  const float* W_v  = (const float*)d_in[9];
  const float* b_v  = (const float*)d_in[10];
  const float* W_o  = (const float*)d_in[11];
  const float* b_o  = (const float*)d_in[12];
  float* out = (float*)d_out;

  const int Mv = BS * NUM_VALUE;   // 119720
  const int Mq = BS * NQ;          // 90000

  char* ws = (char*)d_ws;
  float* vproj = (float*)ws; ws += (size_t)Mv * 256 * sizeof(float);
  float* offb  = (float*)ws; ws += (size_t)Mq * 256 * sizeof(float);
  float* awb   = (float*)ws; ws += (size_t)Mq * 128 * sizeof(float);
  float* resb  = (float*)ws;

  const dim3 blk(256);

  // 1) value projection: (Mv x 256) @ (256 x 256) + b_v
  gemm_wmma_f32<<<dim3((Mv + BLK_M - 1) / BLK_M, 256 / BLK_N), blk, 0,
                  stream>>>(value, W_v, b_v, vproj, Mv, 256);
  // 2) sampling offsets: (Mq x 256) @ (256 x 256) + b_so
  gemm_wmma_f32<<<dim3((Mq + BLK_M - 1) / BLK_M, 256 / BLK_N), blk, 0,
                  stream>>>(query, W_so, b_so, offb, Mq, 256);
  // 3) attention-weight logits: (Mq x 256) @ (256 x 128) + b_aw
  gemm_wmma_f32<<<dim3((Mq + BLK_M - 1) / BLK_M, 128 / BLK_N), blk, 0,
                  stream>>>(query, W_aw, b_aw, awb, Mq, 128);
  // 4) softmax over L*P=16 per (b,q,head)
  softmax16_kernel<<<(Mq * H_HEADS + 255) / 256, blk, 0, stream>>>(
      awb, Mq * H_HEADS);
  // 5) bilinear sampling + point-weighted aggregation
  ms_sample_kernel<<<(Mq * H_HEADS + 7) / 8, blk, 0, stream>>>(
      vproj, offb, awb, refpt, resb);
  // 6) output projection: (Mq x 256) @ (256 x 256) + b_o
  gemm_wmma_f32<<<dim3((Mq + BLK_M - 1) / BLK_M, 256 / BLK_N), blk, 0,
                  stream>>>(resb, W_o, b_o, out, Mq, 256);
}